// GroupSort_SparseConv_558345749013
// MI455X (gfx1250) — compile-verified
//
#include <hip/hip_runtime.h>

// GroupSort over channel pairs of x:(32,512,56,56) fp32, NCHW contiguous.
// Pair (2k,2k+1): z=relu(a-b); out[2k]=a-z (min), out[2k+1]=b+z (max).
// Pure streaming op: 411 MB traffic, ~17.6 us at 23.3 TB/s HBM.
// Strategy: 128-bit coalesced loads/stores with non-temporal hints
// (gfx1250 TH=NT) since the working set (411MB) exceeds the 192MB L2
// and every byte is touched exactly once.

typedef float v4f __attribute__((ext_vector_type(4)));  // native vector: OK for NT builtins

#define HW   3136   // 56*56 floats per plane
#define HW4  784    // float4 per plane

__global__ __launch_bounds__(256)
void groupsort_pair_kernel(const v4f* __restrict__ x,
                           v4f* __restrict__ out,
                           int n4)  // number of even-plane float4 elements
{
    int t = blockIdx.x * blockDim.x + threadIdx.x;
    if (t >= n4) return;

    int p = t / HW4;          // which (even,odd) channel-plane pair
    int i = t - p * HW4;      // float4 offset within the plane

    int base_e = p * (2 * HW4) + i;   // even-channel plane
    int base_o = base_e + HW4;        // odd-channel plane (HW floats later)

    v4f a = __builtin_nontemporal_load(&x[base_e]);
    v4f b = __builtin_nontemporal_load(&x[base_o]);

    v4f e, o, z;
    z.x = fmaxf(a.x - b.x, 0.0f);  e.x = a.x - z.x;  o.x = b.x + z.x;
    z.y = fmaxf(a.y - b.y, 0.0f);  e.y = a.y - z.y;  o.y = b.y + z.y;
    z.z = fmaxf(a.z - b.z, 0.0f);  e.z = a.z - z.z;  o.z = b.z + z.z;
    z.w = fmaxf(a.w - b.w, 0.0f);  e.w = a.w - z.w;  o.w = b.w + z.w;

    __builtin_nontemporal_store(e, &out[base_e]);
    __builtin_nontemporal_store(o, &out[base_o]);
}

extern "C" void kernel_launch(void* const* d_in, const int* in_sizes, int n_in,
                              void* d_out, int out_size, void* d_ws, size_t ws_size,
                              hipStream_t stream)
{
    const v4f* x   = (const v4f*)d_in[0];
    v4f*       out = (v4f*)d_out;

    // total floats = 32*512*56*56 = 51,380,224
    // even-plane float4 count = n / 2 / 4
    int n  = in_sizes[0];
    int n4 = n >> 3;              // 6,422,528

    const int block = 256;
    const int grid  = (n4 + block - 1) / block;   // 25,088 blocks

    groupsort_pair_kernel<<<grid, block, 0, stream>>>(x, out, n4);
}